// LatentGaussianMixture_7937099563215
// MI455X (gfx1250) — compile-verified
//
#include <hip/hip_runtime.h>
#include <math.h>

#define B_ROWS   8192
#define D_DIM    256
#define K_CL     32
#define TILE_ROWS 64
#define LDSS     260                 // padded LDS row stride (float) -> conflict-free column access
#define NBLK     (B_ROWS / TILE_ROWS) // 128

typedef __attribute__((ext_vector_type(2))) float v2f;
typedef __attribute__((ext_vector_type(8))) float v8f;
typedef int b128_t __attribute__((vector_size(16)));   // matches builtin param type

#define AS3 __attribute__((address_space(3)))

#if defined(__has_builtin)
#if __has_builtin(__builtin_amdgcn_global_load_async_to_lds_b128) && \
    __has_builtin(__builtin_amdgcn_global_store_async_from_lds_b128) && \
    __has_builtin(__builtin_amdgcn_s_wait_asynccnt)
#define USE_ASYNC_LDS 1
#endif
#endif
#ifndef USE_ASYNC_LDS
#define USE_ASYNC_LDS 0
#endif

// ws (float) layout:
//   [0,8192)        inv_sig_c            [K,D]
//   [8192,16384)    mu_c * inv_sig_c     [K,D]
//   [16384,16416)   S1[k] = sum_d log_sigma_sq_c
//   [16416,16448)   Q[k]  = sum_d mu_c^2 * inv
//   [16448,16576)   per-block gaussian partial (128)
//   [16576,20672)   per-block sum_pi partial (128*32)
#define WS_INV    0
#define WS_MUINV  8192
#define WS_S1     16384
#define WS_Q      16416
#define WS_OUT    16448

__device__ __forceinline__ v8f wmma4(v2f a, v2f b, v8f c) {
  // D = A(16x4 f32) * B(4x16 f32) + C(16x16 f32)
  return __builtin_amdgcn_wmma_f32_16x16x4_f32(false, a, false, b, (short)0, c,
                                               false, false);
}

__device__ __forceinline__ v8f vz8() {
  v8f r = {0.f, 0.f, 0.f, 0.f, 0.f, 0.f, 0.f, 0.f};
  return r;
}

#if USE_ASYNC_LDS
__device__ __forceinline__ void async_g2l_b128(const void* g, const void* l) {
  // LDS offset = low 32 bits of the generic address (ISA 10.2)
  __builtin_amdgcn_global_load_async_to_lds_b128(
      (b128_t*)g, (AS3 b128_t*)(uint32_t)(uintptr_t)l, 0, 0);
}
__device__ __forceinline__ void async_l2g_b128(void* g, const void* l) {
  __builtin_amdgcn_global_store_async_from_lds_b128(
      (b128_t*)g, (AS3 b128_t*)(uint32_t)(uintptr_t)l, 0, 0);
}
#endif

// ---------------- Phase A: cluster precompute ----------------
__global__ void lgm_prep_kernel(const float* __restrict__ mu_c,
                                const float* __restrict__ logc,
                                float* __restrict__ ws) {
  float* inv   = ws + WS_INV;
  float* muinv = ws + WS_MUINV;
  float* S1    = ws + WS_S1;
  float* Qv    = ws + WS_Q;
  const int tid = threadIdx.x;
  for (int i = tid; i < K_CL * D_DIM; i += blockDim.x) {
    float l  = logc[i];
    float iv = __expf(-l);
    inv[i]   = iv;
    muinv[i] = mu_c[i] * iv;
  }
  __syncthreads();
  if (tid < K_CL) {
    float s1 = 0.f, q = 0.f;
    const float* lr = logc + tid * D_DIM;
    const float* mr = mu_c + tid * D_DIM;
    const float* ir = inv  + tid * D_DIM;
    for (int d = 0; d < D_DIM; ++d) {
      s1 += lr[d];
      q  += mr[d] * mr[d] * ir[d];
    }
    S1[tid] = s1;
    Qv[tid] = q;
  }
}

// ---------------- Phase B: projections + z + logits + loss partials ----------------
__global__ __launch_bounds__(256, 1)
void lgm_main_kernel(const float* __restrict__ x,
                     const float* __restrict__ eps,
                     const float* __restrict__ Wmu,  const float* __restrict__ bmu,
                     const float* __restrict__ Wsig, const float* __restrict__ bsig,
                     const float* __restrict__ ws,
                     float* __restrict__ wsout,
                     float* __restrict__ zout) {
  __shared__ float Xs[TILE_ROWS][LDSS];   // X tile, reused as E = exp(lsz)
  __shared__ float Ls[TILE_ROWS][LDSS];   // lsz, reused as z
  __shared__ float Ms[TILE_ROWS][LDSS];   // mu_z
  __shared__ float Plog[TILE_ROWS][36];   // -(L1 - 2*L2)
  __shared__ float Ploss[TILE_ROWS][36];  // L3 + L4 - 2*L5
  __shared__ float s_wsumpi[8][32];
  __shared__ float s_wgauss[8];

  const int tid  = threadIdx.x;
  const int lane = tid & 31;
  const int wave = tid >> 5;
  const int col  = lane & 15;
  const int hi   = lane >> 4;   // 0 or 1
  const int kh   = hi << 1;     // K sub-offset 0 or 2
  const int row0 = blockIdx.x * TILE_ROWS;

  const float* inv   = ws + WS_INV;
  const float* muinv = ws + WS_MUINV;
  const float* S1    = ws + WS_S1;
  const float* Qv    = ws + WS_Q;
  float* bgauss = wsout;         // [NBLK]
  float* bsumpi = wsout + NBLK;  // [NBLK][32]

  // ---- stage X tile (async global -> LDS when available) ----
  {
    const float4* xg = (const float4*)(x + (size_t)row0 * D_DIM);
#pragma unroll
    for (int t = 0; t < 16; ++t) {
      int idx4 = t * 256 + tid;          // 0..4095
      int r    = idx4 >> 6;
      int c4   = (idx4 & 63) << 2;
#if USE_ASYNC_LDS
      async_g2l_b128(xg + idx4, &Xs[r][c4]);
#else
      *(float4*)&Xs[r][c4] = xg[idx4];
#endif
    }
#if USE_ASYNC_LDS
    __builtin_amdgcn_s_wait_asynccnt(0);
#endif
  }
  __syncthreads();

  // ---- GEMM1: mu_z = X*Wmu^T + bmu ; lsz = X*Wsig^T + bsig (V_WMMA_F32_16X16X4_F32) ----
  // Two independent accumulator chains (even/odd k-groups) for ILP.
  {
    const int mtx = wave >> 2;   // 0 -> mu, 1 -> sigma
    const int rt  = wave & 3;    // row tile 0..3
    const float* W    = mtx ? Wsig : Wmu;
    const float* bias = mtx ? bsig : bmu;
    float* dst0 = mtx ? &Ls[0][0] : &Ms[0][0];
    const int arow = rt * 16 + col;
    for (int n0 = 0; n0 < D_DIM; n0 += 16) {
      if (n0 + 16 < D_DIM) __builtin_prefetch(W + (size_t)(n0 + 16 + col) * D_DIM, 0, 1);
      const float* wrow = W + (size_t)(n0 + col) * D_DIM;  // lane owns output column e=n0+col
      v8f acc0 = vz8(), acc1 = vz8();
#pragma unroll 4
      for (int k0 = 0; k0 < D_DIM; k0 += 8) {
        v2f a0 = *(const v2f*)&Xs[arow][k0 + kh];
        v2f b0 = *(const v2f*)(wrow + k0 + kh);
        v2f a1 = *(const v2f*)&Xs[arow][k0 + 4 + kh];
        v2f b1 = *(const v2f*)(wrow + k0 + 4 + kh);
        acc0 = wmma4(a0, b0, acc0);
        acc1 = wmma4(a1, b1, acc1);
      }
      float bb = bias[n0 + col];
#pragma unroll
      for (int i = 0; i < 8; ++i)
        dst0[(rt * 16 + i + 8 * hi) * LDSS + n0 + col] = acc0[i] + acc1[i] + bb;
    }
  }
  __syncthreads();

  // ---- elementwise: E = exp(lsz); z = mu + sqrt(E)*eps ; stream z out ----
  {
    const float4* eg = (const float4*)(eps + (size_t)row0 * D_DIM);
    float4*       zg = (float4*)(zout + (size_t)row0 * D_DIM);
#pragma unroll
    for (int t = 0; t < 16; ++t) {
      int idx4 = t * 256 + tid;
      int r    = idx4 >> 6;
      int c4   = (idx4 & 63) << 2;
      float4 mu = *(float4*)&Ms[r][c4];
      float4 ls = *(float4*)&Ls[r][c4];
      float4 ep = eg[idx4];
      float4 E, z;
      E.x = __expf(ls.x); E.y = __expf(ls.y); E.z = __expf(ls.z); E.w = __expf(ls.w);
      z.x = mu.x + sqrtf(E.x) * ep.x;
      z.y = mu.y + sqrtf(E.y) * ep.y;
      z.z = mu.z + sqrtf(E.z) * ep.z;
      z.w = mu.w + sqrtf(E.w) * ep.w;
      *(float4*)&Xs[r][c4] = E;   // Xs now holds E
      *(float4*)&Ls[r][c4] = z;   // Ls now holds z
#if USE_ASYNC_LDS
      async_l2g_b128(zg + idx4, &Ls[r][c4]);  // z: LDS -> global, async
#else
      zg[idx4] = z;
#endif
    }
  }
  __syncthreads();

  // ---- GEMM2: five [64x256]x[256x32] WMMA GEMMs, combined in-register ----
  {
    const int rt2 = wave >> 1;            // row tile 0..3
    const int nh  = wave & 1;             // cluster half
    const int arow = rt2 * 16 + col;
    const int ccl  = nh * 16 + col;       // cluster index this lane owns
    const float* invrow = inv   + (size_t)ccl * D_DIM;
    const float* mivrow = muinv + (size_t)ccl * D_DIM;
    v8f a1 = vz8(), a2 = vz8(), a3 = vz8(), a4 = vz8(), a5 = vz8();
#pragma unroll 4
    for (int k0 = 0; k0 < D_DIM; k0 += 4) {
      v2f zv  = *(const v2f*)&Ls[arow][k0 + kh];
      v2f ev  = *(const v2f*)&Xs[arow][k0 + kh];
      v2f mv  = *(const v2f*)&Ms[arow][k0 + kh];
      v2f iv  = *(const v2f*)(invrow + k0 + kh);
      v2f miv = *(const v2f*)(mivrow + k0 + kh);
      v2f z2 = zv * zv;
      v2f m2 = mv * mv;
      a1 = wmma4(z2, iv,  a1);   // z^2   . inv
      a2 = wmma4(zv, miv, a2);   // z     . mu*inv
      a3 = wmma4(ev, iv,  a3);   // E     . inv
      a4 = wmma4(m2, iv,  a4);   // mu^2  . inv
      a5 = wmma4(mv, miv, a5);   // mu    . mu*inv
    }
#pragma unroll
    for (int i = 0; i < 8; ++i) {
      int r = rt2 * 16 + i + 8 * hi;
      Plog[r][nh * 16 + col]  = -(a1[i] - 2.f * a2[i]);
      Ploss[r][nh * 16 + col] = a3[i] + a4[i] - 2.f * a5[i];
    }
  }
  __syncthreads();

  // ---- softmax over K=32 (one wave32 per row, lane = cluster) + loss ----
  {
    const float Qk  = Qv[lane];
    const float S1k = S1[lane];
    float sum_pi = 0.f;
    float wloss  = 0.f;
    for (int rr = 0; rr < 8; ++rr) {
      int r = wave * 8 + rr;
      float lg = Plog[r][lane] - Qk;
      float mx = lg;
#pragma unroll
      for (int off = 16; off >= 1; off >>= 1)
        mx = fmaxf(mx, __shfl_xor(mx, off, 32));
      float e = __expf(lg - mx);
      float s = e;
#pragma unroll
      for (int off = 16; off >= 1; off >>= 1)
        s += __shfl_xor(s, off, 32);
      float pi = e / s + 1e-10f;
      sum_pi += pi;
      float term = pi * (S1k + Qk + Ploss[r][lane]);
#pragma unroll
      for (int off = 16; off >= 1; off >>= 1)
        term += __shfl_xor(term, off, 32);
      if (lane == 0) wloss += term;
    }
    s_wsumpi[wave][lane] = sum_pi;
    if (lane == 0) s_wgauss[wave] = wloss;
  }
  __syncthreads();

  if (tid < 32) {
    float sp = 0.f;
#pragma unroll
    for (int w = 0; w < 8; ++w) sp += s_wsumpi[w][tid];
    bsumpi[blockIdx.x * 32 + tid] = sp;
  } else if (tid == 32) {
    float g = 0.f;
#pragma unroll
    for (int w = 0; w < 8; ++w) g += s_wgauss[w];
    bgauss[blockIdx.x] = g;
  }

#if USE_ASYNC_LDS
  __builtin_amdgcn_s_wait_asynccnt(0);  // drain async z stores before wave exit
#endif
}

// ---------------- Phase C: deterministic final reduction ----------------
__global__ void lgm_finalize_kernel(const float* __restrict__ wsout,
                                    float* __restrict__ out) {
  const int tid = threadIdx.x;  // 32 threads
  const float* bgauss = wsout;
  const float* bsumpi = wsout + NBLK;
  float sp = 0.f;
  for (int b = 0; b < NBLK; ++b) sp += bsumpi[b * 32 + tid];
  float mp = sp * (1.0f / (float)B_ROWS);
  float u  = mp * __logf(mp);
#pragma unroll
  for (int off = 16; off >= 1; off >>= 1) u += __shfl_xor(u, off, 32);
  if (tid == 0) {
    float g = 0.f;
    for (int b = 0; b < NBLK; ++b) g += bgauss[b];
    out[(size_t)B_ROWS * D_DIM]     = 0.5f * g;     // batch_gaussian_loss
    out[(size_t)B_ROWS * D_DIM + 1] = u * (1.0f / (float)K_CL);  // batch_uniform_loss
  }
}

extern "C" void kernel_launch(void* const* d_in, const int* in_sizes, int n_in,
                              void* d_out, int out_size, void* d_ws, size_t ws_size,
                              hipStream_t stream) {
  (void)in_sizes; (void)n_in; (void)out_size; (void)ws_size;
  const float* x    = (const float*)d_in[0];  // embedded_state [B,1,D]
  const float* eps  = (const float*)d_in[1];  // eps            [B,1,D]
  const float* Wmu  = (const float*)d_in[2];  // [D,D]
  const float* bmu  = (const float*)d_in[3];  // [D]
  const float* Wsig = (const float*)d_in[4];  // [D,D]
  const float* bsig = (const float*)d_in[5];  // [D]
  const float* muc  = (const float*)d_in[6];  // [K,D]
  const float* logc = (const float*)d_in[7];  // [K,D]
  float* out = (float*)d_out;                 // z (B*D) then 2 scalars
  float* wsf = (float*)d_ws;

  lgm_prep_kernel<<<1, 256, 0, stream>>>(muc, logc, wsf);
  lgm_main_kernel<<<NBLK, 256, 0, stream>>>(x, eps, Wmu, bmu, Wsig, bsig,
                                            wsf, wsf + WS_OUT, out);
  lgm_finalize_kernel<<<1, 32, 0, stream>>>(wsf + WS_OUT, out);
}